// FraudGNN_71330816852688
// MI455X (gfx1250) — compile-verified
//
#include <hip/hip_runtime.h>

// ---------------------------------------------------------------------------
// FraudGNN (2x GraphSAGE + linear head) for MI455X / gfx1250
//
// Roofline: aggregation traffic ~1.6GB (L2-resident: x/h1/aggr are 25.6MB
// each vs 192MB L2) dominates; GEMMs are ~3.3 GFLOP total. Scatter kernels
// use float4 gathers + global_atomic_add_f32; the dense
// aggr@Wl + h@Wr + b (+ReLU) is fused into one V_WMMA_F32_16X16X4_F32 kernel
// (fp32 WMMA: inputs are fp32 and the GEMM is not the bottleneck).
//
// This revision templates NCOL and hoists all addressing into loop-invariant
// base pointers so in-loop accesses are base[imm] -> global_load immediate
// offsets, eliminating the per-iteration 64-bit address VALU seen in round 1.
// ---------------------------------------------------------------------------

typedef __attribute__((ext_vector_type(2))) float v2f;
typedef __attribute__((ext_vector_type(8))) float v8f;

__global__ void zero_kernel(float* __restrict__ p, size_t n) {
  size_t i = (size_t)blockIdx.x * blockDim.x + threadIdx.x;
  size_t stride = (size_t)gridDim.x * blockDim.x;
  for (; i < n; i += stride) p[i] = 0.0f;
}

__global__ void degree_kernel(const int* __restrict__ dst,
                              float* __restrict__ deg, int E) {
  int e = blockIdx.x * blockDim.x + threadIdx.x;
  if (e < E) unsafeAtomicAdd(&deg[dst[e]], 1.0f);
}

__global__ void invdeg_kernel(float* __restrict__ deg, int n) {
  int i = blockIdx.x * blockDim.x + threadIdx.x;
  if (i < n) deg[i] = 1.0f / fmaxf(deg[i], 1.0f);
}

// One thread per (edge, 4-float chunk); feature width fixed at 64 -> 16 chunks.
// Gather is coalesced along features; scatter via hardware f32 atomics (L2).
__global__ void scatter_kernel(const int* __restrict__ src,
                               const int* __restrict__ dst,
                               const float* __restrict__ feat,
                               float* __restrict__ aggr, int E) {
  int tid = blockIdx.x * blockDim.x + threadIdx.x;
  int e = tid >> 4;
  int c = tid & 15;
  if (e >= E) return;
  int s = src[e];
  int d = dst[e];
  const float4 v = *(const float4*)(feat + (size_t)s * 64 + c * 4);
  float* o = aggr + (size_t)d * 64 + c * 4;
  unsafeAtomicAdd(o + 0, v.x);
  unsafeAtomicAdd(o + 1, v.y);
  unsafeAtomicAdd(o + 2, v.z);
  unsafeAtomicAdd(o + 3, v.w);
}

// Fused SAGE layer: out = act( (aggr * inv_deg) @ Wl + bl + hin @ Wr )
// K = 64 (both layers). One block = 16 rows; one wave per 16-col tile.
// Fragment layouts per CDNA5 ISA (f32 16x16x4):
//   A: lane l holds A[l&15][k0 + 2*(l>>4) + {0,1}]  (float2 load, row-major)
//   B: lane l holds W[kb][l&15], W[kb+1][l&15]       (kb = k0 + 2*(l>>4))
//   C/D: VGPR j -> row j + 8*(l>>4), col l&15
// All lane-varying addressing is hoisted; in-loop indices are compile-time
// constants so they land in the load/store immediate offset fields.
template <int NCOL, bool RELU>
__global__ __launch_bounds__(128) void sage_wmma_kernel(
    const float* __restrict__ aggr, const float* __restrict__ hin,
    const float* __restrict__ inv_deg, const float* __restrict__ Wl,
    const float* __restrict__ bl, const float* __restrict__ Wr,
    float* __restrict__ out, int n) {
  const int lane = threadIdx.x & 31;
  const int wave = threadIdx.x >> 5;
  const int row0 = blockIdx.x * 16;
  const int col0 = wave * 16;
  const int row  = lane & 15;   // A row index
  const int half = lane >> 4;   // K sub-pair select
  const int col  = lane & 15;   // B/D column within tile

  int r = row0 + row;
  if (r >= n) r = n - 1;  // clamp loads; stores predicated (after WMMAs)
  const float s = inv_deg[r];

  // Loop-invariant base pointers (per-lane); loop displacements are constants.
  const float* __restrict__ arow = aggr + (size_t)r * 64 + half * 2;
  const float* __restrict__ hrow = hin + (size_t)r * 64 + half * 2;
  const float* __restrict__ wl0 = Wl + half * 2 * NCOL + col0 + col;
  const float* __restrict__ wr0 = Wr + half * 2 * NCOL + col0 + col;

  v8f acc = {};
#pragma unroll
  for (int k0 = 0; k0 < 64; k0 += 4) {
    v2f aN = *(const v2f*)(arow + k0);  // 8B-aligned (base even)
    aN *= s;  // fold mean-aggregation scale into A fragment
    v2f aR = *(const v2f*)(hrow + k0);
    v2f bL, bR;
    bL.x = wl0[k0 * NCOL];
    bL.y = wl0[(k0 + 1) * NCOL];
    bR.x = wr0[k0 * NCOL];
    bR.y = wr0[(k0 + 1) * NCOL];
    // D = A*B + C ; both paths accumulate into the same C fragment
    acc = __builtin_amdgcn_wmma_f32_16x16x4_f32(false, aN, false, bL,
                                                (short)0, acc, false, false);
    acc = __builtin_amdgcn_wmma_f32_16x16x4_f32(false, aR, false, bR,
                                                (short)0, acc, false, false);
  }

  const float bias = bl[col0 + col];
  const int mbase = row0 + 8 * half;
  float* __restrict__ orow = out + (size_t)mbase * NCOL + col0 + col;
#pragma unroll
  for (int j = 0; j < 8; ++j) {
    if (mbase + j < n) {
      float v = acc[j] + bias;
      if (RELU) v = fmaxf(v, 0.0f);
      orow[j * NCOL] = v;
    }
  }
}

// Final head: N x 32 @ 32 x 2 + bf. Tiny; scalar VALU per node.
__global__ void head_kernel(const float* __restrict__ h2,
                            const float* __restrict__ Wf,
                            const float* __restrict__ bf,
                            float* __restrict__ out, int n) {
  int i = blockIdx.x * blockDim.x + threadIdx.x;
  if (i >= n) return;
  const float* h = h2 + (size_t)i * 32;
  float a0 = bf[0], a1 = bf[1];
#pragma unroll
  for (int k = 0; k < 32; ++k) {
    float v = h[k];
    a0 += v * Wf[k * 2 + 0];
    a1 += v * Wf[k * 2 + 1];
  }
  out[(size_t)i * 2 + 0] = a0;
  out[(size_t)i * 2 + 1] = a1;
}

extern "C" void kernel_launch(void* const* d_in, const int* in_sizes, int n_in,
                              void* d_out, int out_size, void* d_ws,
                              size_t ws_size, hipStream_t stream) {
  const float* x   = (const float*)d_in[0];
  const int*   ei  = (const int*)d_in[1];
  const float* Wl1 = (const float*)d_in[2];
  const float* bl1 = (const float*)d_in[3];
  const float* Wr1 = (const float*)d_in[4];
  const float* Wl2 = (const float*)d_in[5];
  const float* bl2 = (const float*)d_in[6];
  const float* Wr2 = (const float*)d_in[7];
  const float* Wf  = (const float*)d_in[8];
  const float* bf  = (const float*)d_in[9];
  float* out = (float*)d_out;

  const int N = in_sizes[0] / 64;
  const int E = in_sizes[1] / 2;
  const int* src = ei;      // edge_index row 0
  const int* dst = ei + E;  // edge_index row 1

  // Workspace layout (floats): [deg:N][aggr:N*64][h1:N*64][h2:N*32]
  float* ws   = (float*)d_ws;
  float* deg  = ws;
  float* aggr = ws + N;
  float* h1   = aggr + (size_t)N * 64;
  float* h2   = h1 + (size_t)N * 64;

  const int scatter_threads = E * 16;
  const int mt = (N + 15) / 16;

  // deg + aggr are contiguous -> one zeroing pass (re-zeroed every call)
  zero_kernel<<<2048, 256, 0, stream>>>(deg, (size_t)N * 65);
  degree_kernel<<<(E + 255) / 256, 256, 0, stream>>>(dst, deg, E);
  invdeg_kernel<<<(N + 255) / 256, 256, 0, stream>>>(deg, N);

  // Layer 1
  scatter_kernel<<<(scatter_threads + 255) / 256, 256, 0, stream>>>(
      src, dst, x, aggr, E);
  sage_wmma_kernel<64, true><<<mt, 128, 0, stream>>>(aggr, x, deg, Wl1, bl1,
                                                     Wr1, h1, N);

  // Layer 2
  zero_kernel<<<2048, 256, 0, stream>>>(aggr, (size_t)N * 64);
  scatter_kernel<<<(scatter_threads + 255) / 256, 256, 0, stream>>>(
      src, dst, h1, aggr, E);
  sage_wmma_kernel<32, true><<<mt, 64, 0, stream>>>(aggr, h1, deg, Wl2, bl2,
                                                    Wr2, h2, N);

  // Head
  head_kernel<<<(N + 255) / 256, 256, 0, stream>>>(h2, Wf, bf, out, N);
}